// SEP_ICE_45689862094927
// MI455X (gfx1250) — compile-verified
//
#include <hip/hip_runtime.h>

#define N_NODES 50000
#define N_EDGES 800000
#define IN_C    128
#define OUT_C   128
#define EDGE_K  3
#define LRELU_A 0.2f
#define FCOLS   (EDGE_K * OUT_C)   // 384

typedef float v2f __attribute__((ext_vector_type(2)));
typedef float v8f __attribute__((ext_vector_type(8)));

// ---------------------------------------------------------------------------
// float atomic max via ordered-int trick (works with -inf init)
// ---------------------------------------------------------------------------
__device__ __forceinline__ void atomic_max_f32(float* addr, float val) {
    if (val >= 0.0f) {
        atomicMax((int*)addr, __float_as_int(val));
    } else {
        atomicMin((unsigned int*)addr, __float_as_uint(val));
    }
}

// ---------------------------------------------------------------------------
// init: out = 0, m = -inf, denom = 0   (re-run every call: graph-replay safe)
// ---------------------------------------------------------------------------
__global__ void k_init(float* __restrict__ out, float* __restrict__ m,
                       float* __restrict__ den) {
    int i = blockIdx.x * blockDim.x + threadIdx.x;
    if (i < N_NODES * OUT_C) out[i] = 0.0f;
    if (i < N_NODES * EDGE_K) { m[i] = -__builtin_inff(); den[i] = 0.0f; }
}

// ---------------------------------------------------------------------------
// Wf[i, k*128+o] = sum_t W[i,t] * W_out[k*128+t, o]     (128x384 fused weight)
// ---------------------------------------------------------------------------
__global__ void k_fuse_w(const float* __restrict__ W,
                         const float* __restrict__ W_out,
                         float* __restrict__ Wf) {
    int idx = blockIdx.x * blockDim.x + threadIdx.x;
    if (idx >= IN_C * FCOLS) return;
    int i   = idx / FCOLS;
    int col = idx - i * FCOLS;
    int k   = col >> 7;          // branch 0..2
    int o   = col & 127;         // out channel
    float acc = 0.0f;
    for (int t = 0; t < OUT_C; ++t)
        acc += W[i * OUT_C + t] * W_out[(k * OUT_C + t) * OUT_C + o];
    Wf[i * FCOLS + col] = acc;
}

// b1 = W @ a1, b2 = W @ a2  (fused score vectors, s = h @ b)
__global__ void k_fuse_b(const float* __restrict__ W,
                         const float* __restrict__ a1,
                         const float* __restrict__ a2,
                         float* __restrict__ b1, float* __restrict__ b2) {
    int i = threadIdx.x;
    if (i >= IN_C) return;
    float x1 = 0.0f, x2 = 0.0f;
    for (int t = 0; t < OUT_C; ++t) {
        float w = W[i * OUT_C + t];
        x1 += w * a1[t];
        x2 += w * a2[t];
    }
    b1[i] = x1; b2[i] = x2;
}

// ---------------------------------------------------------------------------
// s1[n] = h[n,:]·b1 ; s2[n] = h[n,:]·b2   (one wave32 per node)
// ---------------------------------------------------------------------------
__global__ void k_scores(const float* __restrict__ h,
                         const float* __restrict__ b1,
                         const float* __restrict__ b2,
                         float* __restrict__ s1, float* __restrict__ s2) {
    int wid  = (blockIdx.x * blockDim.x + threadIdx.x) >> 5;
    int lane = threadIdx.x & 31;
    if (wid >= N_NODES) return;
    float4 hv = ((const float4*)(h + (size_t)wid * IN_C))[lane];
    float4 w1 = ((const float4*)b1)[lane];
    float4 w2 = ((const float4*)b2)[lane];
    float t1 = hv.x*w1.x + hv.y*w1.y + hv.z*w1.z + hv.w*w1.w;
    float t2 = hv.x*w2.x + hv.y*w2.y + hv.z*w2.z + hv.w*w2.w;
    #pragma unroll
    for (int off = 16; off > 0; off >>= 1) {
        t1 += __shfl_xor(t1, off, 32);
        t2 += __shfl_xor(t2, off, 32);
    }
    if (lane == 0) { s1[wid] = t1; s2[wid] = t2; }
}

// ---------------------------------------------------------------------------
// Y = h @ Wf   [50000,128] @ [128,384], exact f32 via V_WMMA_F32_16X16X4_F32.
// One wave per 16x16 output tile; 3125*24 = 75000 full waves (EXEC all-1s).
// A 16x4 frag: lane<16 -> (M=lane, K=k0..k0+1); lane>=16 -> (M, K=k0+2..k0+3)
// B 4x16 frag: lane<16 -> (K=k0..k0+1, N=lane); lane>=16 -> (K=k0+2.., N)
// C/D: VGPR r -> M = r + 8*(lane>=16), N = lane&15
// ---------------------------------------------------------------------------
__global__ void k_gemm_y(const float* __restrict__ h,
                         const float* __restrict__ Wf,
                         float* __restrict__ Y) {
    const int TILES_N = FCOLS / 16;      // 24
    const int TILES_M = N_NODES / 16;    // 3125
    int wid  = (blockIdx.x * blockDim.x + threadIdx.x) >> 5;
    int lane = threadIdx.x & 31;
    int tm   = wid / TILES_N;
    int tn   = wid - tm * TILES_N;
    if (tm >= TILES_M) return;           // wave-uniform (grid is exact anyway)

    const int half = lane >> 4;          // 0 | 1
    const int l15  = lane & 15;
    const int koff = half * 2;
    const int row  = tm * 16 + l15;
    const int ncol = tn * 16 + l15;
    const float* __restrict__ hrow = h + (size_t)row * IN_C;

    v8f c = {};
    #pragma unroll
    for (int k0 = 0; k0 < IN_C; k0 += 4) {
        v2f a = *(const v2f*)(hrow + k0 + koff);           // 8B-aligned
        v2f b;
        b.x = Wf[(k0 + koff)     * FCOLS + ncol];
        b.y = Wf[(k0 + koff + 1) * FCOLS + ncol];
        c = __builtin_amdgcn_wmma_f32_16x16x4_f32(
                /*neg_a=*/false, a, /*neg_b=*/false, b,
                /*c_mod=*/(short)0, c, /*reuse_a=*/false, /*reuse_b=*/false);
    }
    #pragma unroll
    for (int r = 0; r < 8; ++r) {
        int mrow = tm * 16 + r + half * 8;
        Y[(size_t)mrow * FCOLS + ncol] = c[r];
    }
}

// ---------------------------------------------------------------------------
// Edge pass A: e = lrelu(s1[src]+s2[dst]); alpha_k = e*ef_k; segment max -> m
// ---------------------------------------------------------------------------
__global__ void k_edge_a(const long long* __restrict__ ei,
                         const float* __restrict__ ef,
                         const float* __restrict__ s1,
                         const float* __restrict__ s2,
                         float* __restrict__ e_ws, float* __restrict__ m) {
    int e = blockIdx.x * blockDim.x + threadIdx.x;
    if (e >= N_EDGES) return;
    int s = (int)ei[e];
    int d = (int)ei[N_EDGES + e];
    float v = s1[s] + s2[d];
    v = (v >= 0.0f) ? v : LRELU_A * v;
    e_ws[e] = v;
    #pragma unroll
    for (int k = 0; k < EDGE_K; ++k)
        atomic_max_f32(&m[d * EDGE_K + k], v * ef[e * EDGE_K + k]);
}

// Edge pass B: ex = exp(alpha - m[dst]); segment sum -> denom
__global__ void k_edge_b(const long long* __restrict__ ei,
                         const float* __restrict__ ef,
                         const float* __restrict__ e_ws,
                         const float* __restrict__ m,
                         float* __restrict__ ex_ws, float* __restrict__ den) {
    int e = blockIdx.x * blockDim.x + threadIdx.x;
    if (e >= N_EDGES) return;
    int d  = (int)ei[N_EDGES + e];
    float v = e_ws[e];
    #pragma unroll
    for (int k = 0; k < EDGE_K; ++k) {
        float ex = __expf(v * ef[e * EDGE_K + k] - m[d * EDGE_K + k]);
        ex_ws[e * EDGE_K + k] = ex;
        unsafeAtomicAdd(&den[d * EDGE_K + k], ex);
    }
}

// Edge pass C: out[dst] += sum_k att_k * Y[src, k*128 : k*128+128]
// one wave32 per edge; lane handles 4 channels (float4). Y is L2-resident.
__global__ void k_edge_c(const long long* __restrict__ ei,
                         const float* __restrict__ ex_ws,
                         const float* __restrict__ den,
                         const float* __restrict__ Y,
                         float* __restrict__ out) {
    int wid  = (blockIdx.x * blockDim.x + threadIdx.x) >> 5;
    int lane = threadIdx.x & 31;
    if (wid >= N_EDGES) return;
    int s = (int)ei[wid];
    int d = (int)ei[N_EDGES + wid];
    float a0 = ex_ws[wid * 3 + 0] / (den[d * 3 + 0] + 1e-16f);
    float a1 = ex_ws[wid * 3 + 1] / (den[d * 3 + 1] + 1e-16f);
    float a2 = ex_ws[wid * 3 + 2] / (den[d * 3 + 2] + 1e-16f);
    const float4* Yv = (const float4*)(Y + (size_t)s * FCOLS);
    float4 y0 = Yv[lane];
    float4 y1 = Yv[32 + lane];
    float4 y2 = Yv[64 + lane];
    float4 acc;
    acc.x = a0 * y0.x + a1 * y1.x + a2 * y2.x;
    acc.y = a0 * y0.y + a1 * y1.y + a2 * y2.y;
    acc.z = a0 * y0.z + a1 * y1.z + a2 * y2.z;
    acc.w = a0 * y0.w + a1 * y1.w + a2 * y2.w;
    float* op = out + (size_t)d * OUT_C + lane * 4;
    unsafeAtomicAdd(op + 0, acc.x);
    unsafeAtomicAdd(op + 1, acc.y);
    unsafeAtomicAdd(op + 2, acc.z);
    unsafeAtomicAdd(op + 3, acc.w);
}

// ---------------------------------------------------------------------------
extern "C" void kernel_launch(void* const* d_in, const int* in_sizes, int n_in,
                              void* d_out, int out_size, void* d_ws, size_t ws_size,
                              hipStream_t stream) {
    const float*     h     = (const float*)d_in[0];
    const long long* ei    = (const long long*)d_in[1];   // int64 edge_index [2,E]
    const float*     ef    = (const float*)d_in[2];
    const float*     W     = (const float*)d_in[3];
    const float*     a1    = (const float*)d_in[4];
    const float*     a2    = (const float*)d_in[5];
    const float*     W_out = (const float*)d_in[6];
    float*           out   = (float*)d_out;

    // workspace carve (~91.4 MB of f32)
    float* p    = (float*)d_ws;
    float* Wf   = p; p += IN_C * FCOLS;            // 49152
    float* b1   = p; p += IN_C;
    float* b2   = p; p += IN_C;
    float* s1   = p; p += N_NODES;
    float* s2   = p; p += N_NODES;
    float* m    = p; p += N_NODES * EDGE_K;
    float* den  = p; p += N_NODES * EDGE_K;
    float* e_ws = p; p += N_EDGES;
    float* ex_ws= p; p += N_EDGES * EDGE_K;
    float* Y    = p; p += (size_t)N_NODES * FCOLS; // 76.8 MB, L2-resident

    k_init  <<<(N_NODES * OUT_C + 255) / 256, 256, 0, stream>>>(out, m, den);
    k_fuse_w<<<(IN_C * FCOLS + 255) / 256,    256, 0, stream>>>(W, W_out, Wf);
    k_fuse_b<<<1, 128, 0, stream>>>(W, a1, a2, b1, b2);
    k_scores<<<N_NODES / 8, 256, 0, stream>>>(h, b1, b2, s1, s2);   // 50000 waves exact
    k_gemm_y<<<(N_NODES / 16) * (FCOLS / 16) / 8, 256, 0, stream>>>(h, Wf, Y); // 9375 blocks
    k_edge_a<<<N_EDGES / 256, 256, 0, stream>>>(ei, ef, s1, s2, e_ws, m);
    k_edge_b<<<N_EDGES / 256, 256, 0, stream>>>(ei, ef, e_ws, m, ex_ws, den);
    k_edge_c<<<N_EDGES / 8, 256, 0, stream>>>(ei, ex_ws, den, Y, out);
}